// TutteLayerDepth_9371618640232
// MI455X (gfx1250) — compile-verified
//
#include <hip/hip_runtime.h>
#include <math.h>

// ---------------- problem constants (match reference) ----------------
#define GG      32
#define NVERT   1024          // GG*GG
#define NEDGE   2945          // 992 + 992 + 961
#define NB      124           // boundary vertices
#define NI      900           // interior vertices
#define NP      912           // NI padded to multiple of 16
#define NT      57            // NP/16 tiles
#define BB      8
#define NSYS    16            // BB * 2 channels
#define NPTSC   500000
#define VPSTRIDE 3069         // NEDGE + NB

// ---------------- workspace layout (floats) ----------------
#define WS_A_OFF   ((size_t)0)
#define WS_RHS_OFF ((size_t)NSYS * NP * NP)                 // 13,307,904
#define WS_BP_OFF  (WS_RHS_OFF + (size_t)NSYS * NP * 2)     // +29,184
#define WS_V_OFF   (WS_BP_OFF + (size_t)NSYS * NB * 2)      // +3,968
// total ~= 53.5 MB — fits in the 192MB L2 during factorization

// ---------------- output layout (floats) ----------------
#define OUT_V1_OFF   ((size_t)BB * NPTSC * 2)               // 8,000,000
#define OUT_DIST_OFF (OUT_V1_OFF + (size_t)BB * NVERT * 2)  // 8,016,384

typedef float v2f __attribute__((ext_vector_type(2)));
typedef float v8f __attribute__((ext_vector_type(8)));

#if __has_builtin(__builtin_amdgcn_wmma_f32_16x16x4_f32)
#define HAVE_WMMA_F32X4 1
#endif

__device__ __forceinline__ float sigmoidf_(float x) {
  return 1.0f / (1.0f + expf(-x));
}

// boundary ordering index: right col, top row reversed, left col reversed, bottom row
__device__ __forceinline__ int bindex(int v) {
  int i = v >> 5, j = v & 31;
  if (j == 31) return i;               // right:  k = i            (32)
  if (i == 31) return 32 + (30 - j);   // top:    j = 30..0        (31)
  if (j == 0)  return 63 + (30 - i);   // left:   i = 30..0        (31)
  return 94 + (j - 1);                 // bottom: j = 1..30        (30)
}

__device__ __forceinline__ bool interior_of(int v, int* ii) {
  int i = v >> 5, j = v & 31;
  *ii = (i - 1) * 30 + (j - 1);
  return (i >= 1 && i <= 30 && j >= 1 && j <= 30);
}

// =====================================================================
// Kernel 1: per (b, channel) system: boundary positions, zero+assemble
// dense interior Laplacian (padded to 912x912) and rhs = -L_IB @ bpos.
// =====================================================================
__global__ void build_system(const float* __restrict__ var_pred,
                             const int*   __restrict__ edges,
                             float* __restrict__ Aall,
                             float* __restrict__ rhsall,
                             float* __restrict__ bposall) {
  const int s = blockIdx.x, b = s >> 1, c = s & 1;
  const int tid = threadIdx.x, nth = blockDim.x;
  __shared__ float ang[NB];
  __shared__ float bpx[NB], bpy[NB];

  float* A   = Aall + (size_t)s * NP * NP;
  float* rhs = rhsall + (size_t)s * NP * 2;
  float* bp  = bposall + (size_t)s * NB * 2;

  // raw boundary angles
  for (int i = tid; i < NB; i += nth) {
    float v = var_pred[((size_t)b * VPSTRIDE + NEDGE + i) * 2 + c];
    ang[i] = sigmoidf_(v * 0.01f) * 0.98f + 0.01f;   // /A_SCALE, EPS_A
  }
  __syncthreads();
  if (tid == 0) {  // tiny serial normalized inclusive cumsum * 2*pi
    float ssum = 0.f;
    for (int i = 0; i < NB; ++i) ssum += ang[i];
    float sc = 6.283185307179586f / ssum, acc = 0.f;
    for (int i = 0; i < NB; ++i) { acc += ang[i]; ang[i] = acc * sc; }
  }
  __syncthreads();
  const float PI = 3.14159265358979323846f;
  for (int i = tid; i < NB; i += nth) {
    float a = ang[i], t = tanf(a), bx, by;
    if (a > 7.f * PI / 4.f || a <= PI / 4.f)      { bx = 1.f;      by = t;    }
    else if (a <= 3.f * PI / 4.f)                 { bx = 1.f / t;  by = 1.f;  }
    else if (a <= 5.f * PI / 4.f)                 { bx = -1.f;     by = -t;   }
    else                                          { bx = -1.f / t; by = -1.f; }
    bpx[i] = bx; bpy[i] = by;
    bp[i * 2 + 0] = bx; bp[i * 2 + 1] = by;
  }

  // zero the padded matrix + rhs, identity on padding diagonal
  float4* A4 = (float4*)A;
  for (size_t i = tid; i < (size_t)NP * NP / 4; i += nth)
    A4[i] = make_float4(0.f, 0.f, 0.f, 0.f);
  for (int i = tid; i < NP * 2; i += nth) rhs[i] = 0.f;
  for (int d = NI + tid; d < NP; d += nth) A[(size_t)d * NP + d] = 1.0f;
  __syncthreads();

  // edge assembly (interior rows only) with float atomics
  for (int e = tid; e < NEDGE; e += nth) {
    int u = edges[e], v = edges[NEDGE + e];
    float wv = var_pred[((size_t)b * VPSTRIDE + e) * 2 + c];
    float w = sigmoidf_(wv * 0.01f) * 0.98f + 0.01f;   // /W_SCALE, EPS
    int iu, iv;
    bool uin = interior_of(u, &iu), vin = interior_of(v, &iv);
    if (uin) {
      atomicAdd(&A[(size_t)iu * NP + iu], w);
      if (vin) atomicAdd(&A[(size_t)iu * NP + iv], -w);
      else {
        int bv = bindex(v);
        atomicAdd(&rhs[iu * 2 + 0], w * bpx[bv]);
        atomicAdd(&rhs[iu * 2 + 1], w * bpy[bv]);
      }
    }
    if (vin) {
      atomicAdd(&A[(size_t)iv * NP + iv], w);
      if (uin) atomicAdd(&A[(size_t)iv * NP + iu], -w);
      else {
        int bu = bindex(u);
        atomicAdd(&rhs[iv * 2 + 0], w * bpx[bu]);
        atomicAdd(&rhs[iv * 2 + 1], w * bpy[bu]);
      }
    }
  }
}

// =====================================================================
// Kernel 2: blocked right-looking LU (no pivoting; L_II is SPD / strictly
// diagonally dominant). One workgroup (32 wave32 waves) per system.
// Trailing Schur update uses V_WMMA_F32_16X16X4_F32 (4 chained WMMAs per
// 16x16x16 tile-pair). Whole 53MB working set is L2-resident.
// =====================================================================
__global__ void __launch_bounds__(1024) lu_factor(float* __restrict__ Aall) {
  float* A = Aall + (size_t)blockIdx.x * NP * NP;
  const int tid  = threadIdx.x;
  const int lane = tid & 31, wid = tid >> 5;
  const int m = lane & 15, half = lane >> 4;
  __shared__ float kk[16][16];

  for (int kt = 0; kt < NT; ++kt) {
    const int k0 = kt * 16;

    // ---- factor 16x16 diagonal block in LDS ----
    if (tid < 256) kk[tid >> 4][tid & 15] = A[(size_t)(k0 + (tid >> 4)) * NP + k0 + (tid & 15)];
    __syncthreads();
    for (int p = 0; p < 16; ++p) {
      if (tid < 16 && tid > p) kk[tid][p] /= kk[p][p];
      __syncthreads();
      if (tid < 256) {
        int r = tid >> 4, cc = tid & 15;
        if (r > p && cc > p) kk[r][cc] -= kk[r][p] * kk[p][cc];
      }
      __syncthreads();
    }
    if (tid < 256) A[(size_t)(k0 + (tid >> 4)) * NP + k0 + (tid & 15)] = kk[tid >> 4][tid & 15];

    const int rem = NP - k0 - 16;

    // ---- L panel: A[i, k-block] <- A[i, k-block] * U_kk^{-1}  (thread per row)
    if (tid < rem) {
      const int gr = k0 + 16 + tid;
      float rv[16];
      #pragma unroll
      for (int p = 0; p < 16; ++p) rv[p] = A[(size_t)gr * NP + k0 + p];
      #pragma unroll
      for (int p = 0; p < 16; ++p) {
        float x = rv[p];
        for (int q = 0; q < p; ++q) x -= rv[q] * kk[q][p];
        rv[p] = x / kk[p][p];
      }
      #pragma unroll
      for (int p = 0; p < 16; ++p) A[(size_t)gr * NP + k0 + p] = rv[p];
    }
    // ---- U panel: A[k-block, j] <- L_kk^{-1} * A[k-block, j]  (thread per col)
    if (tid < rem) {
      const int gc = k0 + 16 + tid;
      float cv[16];
      #pragma unroll
      for (int p = 0; p < 16; ++p) cv[p] = A[(size_t)(k0 + p) * NP + gc];
      #pragma unroll
      for (int p = 0; p < 16; ++p) {
        float x = cv[p];
        for (int q = 0; q < p; ++q) x -= kk[p][q] * cv[q];
        cv[p] = x;
      }
      #pragma unroll
      for (int p = 0; p < 16; ++p) A[(size_t)(k0 + p) * NP + gc] = cv[p];
    }
    __syncthreads();

    // ---- trailing update: C -= L * U, tiles striped over the 32 waves ----
    const int ntt = NT - kt - 1;
    for (int itr = wid; itr < ntt; itr += 32) {          // wave-uniform: EXEC all-ones
      const int i0 = k0 + 16 + itr * 16;
      // A-fragment (16x4 f32 layout, ISA 7.12.2): lanes 0-15 K={0,1}, 16-31 K={2,3}
      v2f af[4];
      #pragma unroll
      for (int kq = 0; kq < 4; ++kq)
        af[kq] = *(const v2f*)&A[(size_t)(i0 + m) * NP + k0 + kq * 4 + half * 2];

      for (int jtr = 0; jtr < ntt; ++jtr) {
        const int j0 = k0 + 16 + jtr * 16;
#ifdef HAVE_WMMA_F32X4
        v8f c;
        #pragma unroll
        for (int r = 0; r < 8; ++r)                      // C/D layout: VGPR r = rows r / r+8
          c[r] = -A[(size_t)(i0 + r + half * 8) * NP + j0 + m];
        #pragma unroll
        for (int kq = 0; kq < 4; ++kq) {
          const int kb = k0 + kq * 4 + half * 2;
          v2f bf;                                        // B 4x16: VGPR0 K={0,2}, VGPR1 K={1,3}
          bf[0] = A[(size_t)kb * NP + j0 + m];
          bf[1] = A[(size_t)(kb + 1) * NP + j0 + m];
          // acc = L*U + (-C)  (f32 WMMA has no A/B negate; C-negate done in VALU)
          c = __builtin_amdgcn_wmma_f32_16x16x4_f32(false, af[kq], false, bf,
                                                    (short)0, c, false, false);
        }
        #pragma unroll
        for (int r = 0; r < 8; ++r)
          A[(size_t)(i0 + r + half * 8) * NP + j0 + m] = -c[r];
#else
        // scalar fallback (same math, no fragments)
        #pragma unroll
        for (int r = 0; r < 8; ++r) {
          const int row = i0 + r + half * 8, col = j0 + m;
          float acc = A[(size_t)row * NP + col];
          for (int q = 0; q < 16; ++q)
            acc -= A[(size_t)row * NP + k0 + q] * A[(size_t)(k0 + q) * NP + col];
          A[(size_t)row * NP + col] = acc;
        }
#endif
      }
    }
    __syncthreads();
  }
}

// =====================================================================
// Kernel 3: triangular solves (2 RHS, right-looking, LDS-resident) and
// scatter of the embedding into ws V; channel 0 also writes the V1 output.
// =====================================================================
__global__ void __launch_bounds__(1024) solve_scatter(const float* __restrict__ Aall,
                                                      const float* __restrict__ rhsall,
                                                      const float* __restrict__ bposall,
                                                      float* __restrict__ Vall,
                                                      float* __restrict__ out) {
  const int s = blockIdx.x, b = s >> 1, c = s & 1;
  const float* A   = Aall + (size_t)s * NP * NP;
  const float* rhs = rhsall + (size_t)s * NP * 2;
  const float* bp  = bposall + (size_t)s * NB * 2;
  float* V = Vall + (size_t)s * NVERT * 2;
  const int tid = threadIdx.x;
  __shared__ float y0[NI], y1[NI];

  if (tid < NI) { y0[tid] = rhs[tid * 2 + 0]; y1[tid] = rhs[tid * 2 + 1]; }
  __syncthreads();

  // forward: L (unit lower) y = rhs
  for (int j = 0; j < NI - 1; ++j) {
    float yj0 = y0[j], yj1 = y1[j];
    if (tid > j && tid < NI) {
      float l = A[(size_t)tid * NP + j];
      y0[tid] -= l * yj0; y1[tid] -= l * yj1;
    }
    __syncthreads();
  }
  // backward: U x = y
  for (int j = NI - 1; j >= 0; --j) {
    if (tid == 0) { float d = A[(size_t)j * NP + j]; y0[j] /= d; y1[j] /= d; }
    __syncthreads();
    float xj0 = y0[j], xj1 = y1[j];
    if (tid < j) {
      float u = A[(size_t)tid * NP + j];
      y0[tid] -= u * xj0; y1[tid] -= u * xj1;
    }
    __syncthreads();
  }

  // scatter into V (and V1 output for channel 0)
  for (int v = tid; v < NVERT; v += blockDim.x) {
    int ii; float vx, vy;
    if (interior_of(v, &ii)) { vx = y0[ii]; vy = y1[ii]; }
    else { int bi = bindex(v); vx = bp[bi * 2 + 0]; vy = bp[bi * 2 + 1]; }
    V[v * 2 + 0] = vx; V[v * 2 + 1] = vy;
    if (c == 0) {
      out[OUT_V1_OFF + ((size_t)b * NVERT + v) * 2 + 0] = vx;
      out[OUT_V1_OFF + ((size_t)b * NVERT + v) * 2 + 1] = vy;
    }
  }
}

// =====================================================================
// Kernel 4: streaming interpolation over 4M points (HBM-bound, ~10us).
// =====================================================================
__global__ void __launch_bounds__(256) interp_points(const float* __restrict__ pts,
                                                     const float* __restrict__ depth,
                                                     const float* __restrict__ verts,
                                                     const int*   __restrict__ faces,
                                                     const int*   __restrict__ face_ids,
                                                     const float* __restrict__ Vall,
                                                     float* __restrict__ out) {
  unsigned int g = blockIdx.x * blockDim.x + threadIdx.x;
  if (g >= (unsigned int)(BB * NPTSC)) return;
  const unsigned int b = g / NPTSC;

  float px = pts[(size_t)g * 2 + 0], py = pts[(size_t)g * 2 + 1];
  float dep = depth[g];
  int f = face_ids[g];
  int t0 = faces[f * 3 + 0], t1 = faces[f * 3 + 1], t2 = faces[f * 3 + 2];
  float P0x = verts[t0 * 2], P0y = verts[t0 * 2 + 1];
  float P1x = verts[t1 * 2], P1y = verts[t1 * 2 + 1];
  float P2x = verts[t2 * 2], P2y = verts[t2 * 2 + 1];

  float d0x = px - P0x, d0y = py - P0y;
  float d1x = px - P1x, d1y = py - P1y;
  float d2x = px - P2x, d2y = py - P2y;
  float aA = 0.5f * fabsf(d1x * d2y - d1y * d2x);
  float aB = 0.5f * fabsf(d0x * d2y - d0y * d2x);
  float aC = 0.5f * fabsf(d0x * d1y - d0y * d1x);
  float inv = 1.0f / (aA + aB + aC);
  float w0 = aA * inv, w1 = aB * inv, w2 = aC * inv;

  float e1x = P1x - P0x, e1y = P1y - P0y;
  float e2x = P2x - P0x, e2y = P2y - P0y;
  float idet = 1.0f / (e1x * e2y - e1y * e2x);

  const float* V1 = Vall + (size_t)(b * 2 + 0) * NVERT * 2;
  const float* V2 = Vall + (size_t)(b * 2 + 1) * NVERT * 2;

  float prx[2], pry[2], J00[2], J01[2], J10[2], J11[2];
  #pragma unroll
  for (int ch = 0; ch < 2; ++ch) {
    const float* V = ch ? V2 : V1;
    float q0x = V[t0 * 2], q0y = V[t0 * 2 + 1];
    float q1x = V[t1 * 2], q1y = V[t1 * 2 + 1];
    float q2x = V[t2 * 2], q2y = V[t2 * 2 + 1];
    prx[ch] = w0 * q0x + w1 * q1x + w2 * q2x;
    pry[ch] = w0 * q0y + w1 * q1y + w2 * q2y;
    float f1x = q1x - q0x, f1y = q1y - q0y;
    float f2x = q2x - q0x, f2y = q2y - q0y;
    // J = Eq @ invEp,  invEp = [[e2y,-e2x],[-e1y,e1x]] * idet
    J00[ch] = (f1x * e2y - f2x * e1y) * idet;
    J01[ch] = (f2x * e1x - f1x * e2x) * idet;
    J10[ch] = (f1y * e2y - f2y * e1y) * idet;
    J11[ch] = (f2y * e1x - f1y * e2x) * idet;
  }

  float wa = 0.5f * (1.0f - dep), wb = 0.5f * (1.0f + dep);
  out[(size_t)g * 2 + 0] = prx[0] * wa + prx[1] * wb;
  out[(size_t)g * 2 + 1] = pry[0] * wa + pry[1] * wb;
  size_t db = OUT_DIST_OFF + (size_t)g * 4;
  out[db + 0] = J00[0] * wa + J00[1] * wb;
  out[db + 1] = J01[0] * wa + J01[1] * wb;
  out[db + 2] = J10[0] * wa + J10[1] * wb;
  out[db + 3] = J11[0] * wa + J11[1] * wb;
}

// =====================================================================
extern "C" void kernel_launch(void* const* d_in, const int* in_sizes, int n_in,
                              void* d_out, int out_size, void* d_ws, size_t ws_size,
                              hipStream_t stream) {
  const float* pts      = (const float*)d_in[0];  // (B, NPTS, 2)
  const float* var_pred = (const float*)d_in[1];  // (B, E+nB, 2)
  const float* depth    = (const float*)d_in[2];  // (B, NPTS)
  const float* vertices = (const float*)d_in[3];  // (1024, 2)
  const int*   edges    = (const int*)d_in[4];    // (2, E)
  const int*   faces    = (const int*)d_in[5];    // (1922, 3)
  const int*   face_ids = (const int*)d_in[6];    // (B, NPTS)
  float* out = (float*)d_out;

  float* ws   = (float*)d_ws;                     // needs ~53.5 MB
  float* A    = ws + WS_A_OFF;
  float* rhs  = ws + WS_RHS_OFF;
  float* bpos = ws + WS_BP_OFF;
  float* V    = ws + WS_V_OFF;

  build_system<<<NSYS, 256, 0, stream>>>(var_pred, edges, A, rhs, bpos);
  lu_factor<<<NSYS, 1024, 0, stream>>>(A);
  solve_scatter<<<NSYS, 1024, 0, stream>>>(A, rhs, bpos, V, out);

  const int npts_total = BB * NPTSC;
  interp_points<<<(npts_total + 255) / 256, 256, 0, stream>>>(
      pts, depth, vertices, faces, face_ids, V, out);
}